// ContrastiveKernelLoss_21208548507918
// MI455X (gfx1250) — compile-verified
//
#include <hip/hip_runtime.h>
#include <hip/hip_bf16.h>
#include <math.h>

typedef __attribute__((ext_vector_type(2))) float v2f;
typedef __attribute__((ext_vector_type(8))) float v8f;

#define EPS 1e-8f

// ---------------------------------------------------------------------------
// Feature construction: one thread per d x d kernel matrix.
//   u_i = [ -2*vec(inv_i), vec(inv_i^T inv_i), 1, 0-pad ]   (row i of U, n x KP)
//   v_j = [ vec(kn_j^T),   vec(kn_j kn_j^T),  d, 0-pad ]   (col j of V, KP x n)
// so that  <u_i, v_j> = d - 2*tr(inv_i kn_j) + ||inv_i kn_j||_F^2
//                     = ||I - inv_i kn_j||_F^2
// ---------------------------------------------------------------------------
template <int D, int KP>
__global__ void feat_kernel(const float* __restrict__ K,
                            float* __restrict__ U,
                            float* __restrict__ V, int n) {
  int i = blockIdx.x * blockDim.x + threadIdx.x;
  if (i >= n) return;

  float m[D][D];
  float ss = 0.f;
#pragma unroll
  for (int r = 0; r < D; ++r)
#pragma unroll
    for (int c = 0; c < D; ++c) {
      float v = K[(size_t)i * (D * D) + r * D + c];
      m[r][c] = v;
      ss += v * v;
    }
  float inorm = 1.f / (sqrtf(ss) + EPS);
#pragma unroll
  for (int r = 0; r < D; ++r)
#pragma unroll
    for (int c = 0; c < D; ++c) m[r][c] *= inorm;

  float inv[D][D];
  if constexpr (D == 3) {
    float det = m[0][0] * (m[1][1] * m[2][2] - m[1][2] * m[2][1])
              - m[0][1] * (m[1][0] * m[2][2] - m[1][2] * m[2][0])
              + m[0][2] * (m[1][0] * m[2][1] - m[1][1] * m[2][0]);
    float id = 1.f / det;
    inv[0][0] = (m[1][1] * m[2][2] - m[1][2] * m[2][1]) * id;
    inv[0][1] = (m[0][2] * m[2][1] - m[0][1] * m[2][2]) * id;
    inv[0][2] = (m[0][1] * m[1][2] - m[0][2] * m[1][1]) * id;
    inv[1][0] = (m[1][2] * m[2][0] - m[1][0] * m[2][2]) * id;
    inv[1][1] = (m[0][0] * m[2][2] - m[0][2] * m[2][0]) * id;
    inv[1][2] = (m[0][2] * m[1][0] - m[0][0] * m[1][2]) * id;
    inv[2][0] = (m[1][0] * m[2][1] - m[1][1] * m[2][0]) * id;
    inv[2][1] = (m[0][1] * m[2][0] - m[0][0] * m[2][1]) * id;
    inv[2][2] = (m[0][0] * m[1][1] - m[0][1] * m[1][0]) * id;
  } else {
    // Fully unrolled Gauss-Jordan on augmented [A | I]; constant indices after
    // unroll -> stays in VGPRs. Normalized random Gaussian matrices are a.s.
    // invertible without pivoting.
    float a[D][2 * D];
#pragma unroll
    for (int r = 0; r < D; ++r)
#pragma unroll
      for (int c = 0; c < D; ++c) {
        a[r][c] = m[r][c];
        a[r][D + c] = (r == c) ? 1.f : 0.f;
      }
#pragma unroll
    for (int p = 0; p < D; ++p) {
      float piv = 1.f / a[p][p];
#pragma unroll
      for (int c = 0; c < 2 * D; ++c) a[p][c] *= piv;
#pragma unroll
      for (int r = 0; r < D; ++r) {
        if (r == p) continue;
        float f = a[r][p];
#pragma unroll
        for (int c = 0; c < 2 * D; ++c) a[r][c] -= f * a[p][c];
      }
    }
#pragma unroll
    for (int r = 0; r < D; ++r)
#pragma unroll
      for (int c = 0; c < D; ++c) inv[r][c] = a[r][D + c];
  }

  float u[KP], vc[KP];
#pragma unroll
  for (int k = 0; k < KP; ++k) { u[k] = 0.f; vc[k] = 0.f; }
#pragma unroll
  for (int a = 0; a < D; ++a)
#pragma unroll
    for (int b = 0; b < D; ++b) {
      u[a * D + b]  = -2.f * inv[a][b];
      vc[a * D + b] = m[b][a];                 // vec(kn^T)
      float p = 0.f, q = 0.f;
#pragma unroll
      for (int c = 0; c < D; ++c) {
        p += inv[c][a] * inv[c][b];            // (inv^T inv)[a][b]
        q += m[a][c] * m[b][c];                // (kn kn^T)[a][b]
      }
      u[D * D + a * D + b]  = p;
      vc[D * D + a * D + b] = q;
    }
  u[2 * D * D]  = 1.f;
  vc[2 * D * D] = (float)D;

#pragma unroll
  for (int k = 0; k < KP; ++k) {
    U[(size_t)i * KP + k] = u[k];       // row-major, KP is a multiple of 4
    V[(size_t)k * n + i]  = vc[k];      // K-major: column j contiguous per k
  }
}

// ---------------------------------------------------------------------------
// Pairwise GEMM via V_WMMA_F32_16X16X4_F32.
// 256-thread block = 8 wave32; wave w (= threadIdx.y, uniform per wave since
// blockDim.x == 32) owns one 16x16 tile of D2[i,j] = <u_i, v_j>.
// A 16x4 layout : lane L -> M = L%16 ; VGPR0 = K = 2*(L>=16), VGPR1 = K+1
// B  4x16 layout: lane L -> N = L%16 ; VGPR0 = K = 2*(L>=16), VGPR1 = K+1
// C/D 16x16     : lane L -> N = L%16 ; VGPR v -> M = v + 8*(L>=16)
// ---------------------------------------------------------------------------
template <int KP>
__global__ void pair_gemm(const float* __restrict__ U,
                          const float* __restrict__ V, int n, float scale,
                          float* __restrict__ out) {
  int ntile = n >> 4;
  int ti = blockIdx.y;
  int tj = blockIdx.x * 8 + threadIdx.y;   // wave-uniform
  if (tj >= ntile || ti < tj) return;      // whole-wave exit -> EXEC all-1s

  int lane = threadIdx.x;        // 0..31
  int half = lane >> 4;          // 0 or 1
  int l16  = lane & 15;
  int i = ti * 16 + l16;         // M row this lane feeds for A
  int j = tj * 16 + l16;         // N col this lane feeds for B

  v8f acc = {};
#pragma unroll
  for (int kk = 0; kk < KP; kk += 4) {
    int ka = kk + half * 2;
    v2f a, b;
    a.x = U[(size_t)i * KP + ka];
    a.y = U[(size_t)i * KP + ka + 1];
    b.x = V[(size_t)ka * n + j];
    b.y = V[(size_t)(ka + 1) * n + j];
    // (neg_a, A, neg_b, B, c_mod, C, reuse_a, reuse_b)
    acc = __builtin_amdgcn_wmma_f32_16x16x4_f32(false, a, false, b, (short)0,
                                                acc, false, false);
  }

  // acc[v] = D^2 at (I = ti*16 + v + 8*half, J = tj*16 + l16)
  // Branchless epilogue: hardware v_sqrt_f32 (~1 ulp) + float mask, no EXEC
  // manipulation per element.
  float sum = 0.f;
#pragma unroll
  for (int v = 0; v < 8; ++v) {
    int I = ti * 16 + v + half * 8;
    int J = tj * 16 + l16;
    float dd = __builtin_amdgcn_sqrtf(fmaxf(acc[v], 0.f));
    float r  = fmaxf(1.0f - dd, 0.f);        // relu(MARGIN - diff)
    sum += (I > J) ? r : 0.f;                // strict lower triangle only
  }
#pragma unroll
  for (int off = 16; off > 0; off >>= 1) sum += __shfl_xor(sum, off, 32);
  if (lane == 0) atomicAdd(out, sum * scale);
}

__global__ void zero_out(float* out) {
  if (threadIdx.x == 0 && blockIdx.x == 0) out[0] = 0.f;
}

extern "C" void kernel_launch(void* const* d_in, const int* in_sizes, int n_in,
                              void* d_out, int out_size, void* d_ws,
                              size_t ws_size, hipStream_t stream) {
  const float* k0 = (const float*)d_in[0];  // (2048, 3, 3)
  const float* k1 = (const float*)d_in[1];  // (1024, 5, 5)
  float* out = (float*)d_out;

  const int n0 = in_sizes[0] / 9;    // 2048
  const int n1 = in_sizes[1] / 25;   // 1024
  const int KP0 = 20;                // 2*9+1 -> pad to 20
  const int KP1 = 52;                // 2*25+1 -> pad to 52

  float* ws = (float*)d_ws;
  float* U0 = ws;
  float* V0 = U0 + (size_t)n0 * KP0;
  float* U1 = V0 + (size_t)n0 * KP0;
  float* V1 = U1 + (size_t)n1 * KP1;

  zero_out<<<1, 32, 0, stream>>>(out);

  feat_kernel<3, KP0><<<(n0 + 255) / 256, 256, 0, stream>>>(k0, U0, V0, n0);
  feat_kernel<5, KP1><<<(n1 + 255) / 256, 256, 0, stream>>>(k1, U1, V1, n1);

  // loss = (L0 + L1)/2, L = 2*S/(n(n-1))  ->  each layer contributes S/(n(n-1))
  const int nt0 = n0 / 16, nt1 = n1 / 16;
  dim3 blk(32, 8);
  dim3 g0((nt0 + 7) / 8, nt0);
  dim3 g1((nt1 + 7) / 8, nt1);
  pair_gemm<KP0><<<g0, blk, 0, stream>>>(U0, V0, n0,
                                         1.0f / ((float)n0 * (n0 - 1)), out);
  pair_gemm<KP1><<<g1, blk, 0, stream>>>(U1, V1, n1,
                                         1.0f / ((float)n1 * (n1 - 1)), out);
}